// GNN_ADD_24249385353481
// MI455X (gfx1250) — compile-verified
//
#include <hip/hip_runtime.h>

typedef float v2f __attribute__((ext_vector_type(2)));
typedef float v8f __attribute__((ext_vector_type(8)));

// ---------------- degree / normalization ----------------

__global__ void zero_f32_kernel(float* __restrict__ p, int n) {
    int i = blockIdx.x * blockDim.x + threadIdx.x;
    if (i < n) p[i] = 0.0f;
}

__global__ void count_deg_kernel(const int* __restrict__ dst, float* __restrict__ deg, int E) {
    int i = blockIdx.x * blockDim.x + threadIdx.x;
    if (i < E) atomicAdd(&deg[dst[i]], 1.0f);
}

__global__ void make_dinv_kernel(float* __restrict__ dinv, float* __restrict__ dinv2, int n) {
    int i = blockIdx.x * blockDim.x + threadIdx.x;
    if (i < n) {
        float d = dinv[i] + 1.0f;          // +1 self loop; deg integer-exact in f32
        dinv[i]  = 1.0f / sqrtf(d);
        dinv2[i] = 1.0f / d;
    }
}

// ---------------- dense GEMM via V_WMMA_F32_16X16X4_F32 ----------------
// One wave computes a full 16(M) x FO strip of H = X[N,K] @ W[K,FO].
// A 16x4 f32 layout: lanes 0-15 hold K={k0,k0+1} (VGPR0/1), lanes 16-31 hold
// K={k0+2,k0+3}; M = lane&15. B 4x16 mirrors with N = lane&15. D: VGPR r ->
// row m0 + r + 8*(lane>>4), col = lane&15 within tile.
// The A fragment is loaded once per K-step and reused across all FO/16 B tiles.
template<int K, int FO>
__global__ void gemm_wmma_kernel(const float* __restrict__ X, const float* __restrict__ W,
                                 float* __restrict__ Hout, int nTilesM) {
    constexpr int NT = FO / 16;
    int wave = (int)((blockIdx.x * blockDim.x + threadIdx.x) >> 5);
    if (wave >= nTilesM) return;            // wave-uniform exit (EXEC all-1s at WMMA)

    const int m0   = wave << 4;
    const int lane = threadIdx.x & 31;
    const int half = lane >> 4;             // 0 or 1
    const int idx  = lane & 15;
    const int row  = m0 + idx;

    v8f acc[NT];
#pragma unroll
    for (int t = 0; t < NT; ++t) acc[t] = (v8f){};

    for (int k0 = 0; k0 < K; k0 += 4) {
        const int ka = k0 + 2 * half;       // < K whenever K % 4 == 0
        v2f a;
        if constexpr (K % 4 == 0) {
            a = *(const v2f*)(X + (size_t)row * K + ka);
        } else {                            // K == 2: branchless lane select
            v2f av = *(const v2f*)(X + (size_t)row * K);
            a.x = half ? 0.0f : av.x;
            a.y = half ? 0.0f : av.y;
        }
#pragma unroll
        for (int t = 0; t < NT; ++t) {
            const int col = (t << 4) + idx;
            v2f b;
            if constexpr (K % 4 == 0) {
                b.x = W[(size_t)ka * FO + col];
                b.y = W[(size_t)(ka + 1) * FO + col];
            } else {                        // K == 2
                float b0 = W[col];
                float b1 = W[FO + col];
                b.x = half ? 0.0f : b0;
                b.y = half ? 0.0f : b1;
            }
            acc[t] = __builtin_amdgcn_wmma_f32_16x16x4_f32(
                /*neg_a=*/false, a, /*neg_b=*/false, b,
                /*c_mod=*/(short)0, acc[t], /*reuse_a=*/false, /*reuse_b=*/false);
        }
    }

#pragma unroll
    for (int t = 0; t < NT; ++t) {
        const int col = (t << 4) + idx;
#pragma unroll
        for (int r = 0; r < 8; ++r)
            Hout[(size_t)(m0 + r + 8 * half) * FO + col] = acc[t][r];
    }
}

// scalar tail for n % 16 rows (not launched when N is a multiple of 16)
__global__ void gemm_tail_kernel(const float* __restrict__ X, const float* __restrict__ W,
                                 float* __restrict__ Hout, int rowStart, int n, int K, int Fo) {
    int t = blockIdx.x * blockDim.x + threadIdx.x;
    int rows = n - rowStart;
    if (t >= rows * Fo) return;
    int r = rowStart + t / Fo, c = t % Fo;
    float acc = 0.0f;
    for (int k = 0; k < K; ++k) acc += X[(size_t)r * K + k] * W[(size_t)k * Fo + c];
    Hout[(size_t)r * Fo + c] = acc;
}

// ---------------- aggregation ----------------

// agg[i,f] = h[i,f] * dinv2[i]  (self-loop term), float4 vectorized
template<int F>
__global__ void agg_init_kernel(const float* __restrict__ h, const float* __restrict__ dinv2,
                                float* __restrict__ agg, int n) {
    constexpr int Fg = F / 4;
    int t = blockIdx.x * blockDim.x + threadIdx.x;
    if (t >= n * Fg) return;
    int i  = t / Fg;                        // power-of-two -> shift
    int f4 = (t & (Fg - 1)) << 2;
    float  s  = dinv2[i];
    float4 hv = *(const float4*)(h + (size_t)i * F + f4);
    float4 o;
    o.x = hv.x * s; o.y = hv.y * s; o.z = hv.z * s; o.w = hv.w * s;
    *(float4*)(agg + (size_t)i * F + f4) = o;
}

// scatter-add: consecutive threads = consecutive float4 chunks of one edge ->
// coalesced h[src] loads and coalesced f32 atomics into one dst row (L2-resident)
template<int F>
__global__ void edge_scatter_kernel(const int* __restrict__ src, const int* __restrict__ dst,
                                    const float* __restrict__ dinv, const float* __restrict__ h,
                                    float* __restrict__ agg, int E) {
    constexpr int Fg = F / 4;
    long long t = (long long)blockIdx.x * blockDim.x + threadIdx.x;
    if (t >= (long long)E * Fg) return;
    int e  = (int)(t >> __builtin_ctz(Fg)); // power-of-two -> shift
    int f4 = ((int)t & (Fg - 1)) << 2;
    int s  = src[e];
    int d  = dst[e];
    float nrm = dinv[s] * dinv[d];
    float4 hv = *(const float4*)(h + (size_t)s * F + f4);
    float* ap = agg + (size_t)d * F + f4;
    atomicAdd(ap + 0, hv.x * nrm);
    atomicAdd(ap + 1, hv.y * nrm);
    atomicAdd(ap + 2, hv.z * nrm);
    atomicAdd(ap + 3, hv.w * nrm);
}

// x[i,f] = relu(agg[i,f] + b[f])  (in place)
__global__ void bias_relu_kernel(float* __restrict__ x, const float* __restrict__ b,
                                 int n, int F) {
    int t = blockIdx.x * blockDim.x + threadIdx.x;
    if (t >= n * F) return;
    int f = t & (F - 1);                    // F is a power of two (128/32/64)
    float v = x[t] + b[f];
    x[t] = v > 0.0f ? v : 0.0f;
}

// out[i] = x[i,:] . Wl + bl  (no relu on the final linear)
__global__ void final_linear_kernel(const float* __restrict__ x, const float* __restrict__ Wl,
                                    const float* __restrict__ bl, float* __restrict__ out,
                                    int n, int F) {
    int i = blockIdx.x * blockDim.x + threadIdx.x;
    if (i >= n) return;
    float acc = bl[0];
    for (int f = 0; f < F; ++f) acc += x[(size_t)i * F + f] * Wl[f];
    out[i] = acc;
}

// ---------------- per-layer driver ----------------

template<int K, int FO>
static void run_layer(const float* X, const float* W, const float* b,
                      const int* srcp, const int* dstp,
                      const float* dinv, const float* dinv2,
                      float* bufH, float* bufX, int N, int E, hipStream_t stream) {
    const int TB = 256;

    // h = X @ W  (one wave per 16-row strip, 8 waves per block)
    int nTilesM = N / 16;
    int blocks  = (nTilesM * 32 + TB - 1) / TB;
    gemm_wmma_kernel<K, FO><<<blocks, TB, 0, stream>>>(X, W, bufH, nTilesM);
    int rem = N & 15;
    if (rem) {
        int tot = rem * FO;
        gemm_tail_kernel<<<(tot + TB - 1) / TB, TB, 0, stream>>>(X, W, bufH,
                                                                 nTilesM * 16, N, K, FO);
    }

    // agg = h * dinv2  (self loop)
    int tot1 = N * (FO / 4);
    agg_init_kernel<FO><<<(tot1 + TB - 1) / TB, TB, 0, stream>>>(bufH, dinv2, bufX, N);

    // agg[dst] += h[src] * dinv[src]*dinv[dst]
    long long tot2 = (long long)E * (FO / 4);
    edge_scatter_kernel<FO><<<(int)((tot2 + TB - 1) / TB), TB, 0, stream>>>(
        srcp, dstp, dinv, bufH, bufX, E);

    // x = relu(agg + b)
    int tot3 = N * FO;
    bias_relu_kernel<<<(tot3 + TB - 1) / TB, TB, 0, stream>>>(bufX, b, N, FO);
}

// ---------------- entry point ----------------

extern "C" void kernel_launch(void* const* d_in, const int* in_sizes, int n_in,
                              void* d_out, int out_size, void* d_ws, size_t ws_size,
                              hipStream_t stream) {
    (void)n_in; (void)out_size; (void)ws_size;
    const float* x_in = (const float*)d_in[0];
    const int*   edge = (const int*)d_in[1];
    const float* W1 = (const float*)d_in[3];  const float* b1 = (const float*)d_in[4];
    const float* W2 = (const float*)d_in[5];  const float* b2 = (const float*)d_in[6];
    const float* W3 = (const float*)d_in[7];  const float* b3 = (const float*)d_in[8];
    const float* W4 = (const float*)d_in[9];  const float* b4 = (const float*)d_in[10];
    const float* W5 = (const float*)d_in[11]; const float* b5 = (const float*)d_in[12];
    const float* Wl = (const float*)d_in[13]; const float* bl = (const float*)d_in[14];

    const int N  = in_sizes[0] / 2;   // x is [N, 2]
    const int E  = in_sizes[1] / 2;   // edge_index is [2, E]
    const int Hd = 128;               // hidden dims fixed by the reference
    const int Od = 64;

    const int* srcp = edge;
    const int* dstp = edge + E;

    // workspace: bufH | bufX | dinv | dinv2   (~103 MB)
    float* bufH  = (float*)d_ws;
    float* bufX  = bufH + (size_t)N * Hd;
    float* dinv  = bufX + (size_t)N * Hd;
    float* dinv2 = dinv + N;

    const int TB = 256;

    // normalization (deg accumulated in dinv buffer, transformed in place)
    zero_f32_kernel<<<(N + TB - 1) / TB, TB, 0, stream>>>(dinv, N);
    count_deg_kernel<<<(E + TB - 1) / TB, TB, 0, stream>>>(dstp, dinv, E);
    make_dinv_kernel<<<(N + TB - 1) / TB, TB, 0, stream>>>(dinv, dinv2, N);

    // 5 GCN layers: 2->128->128->32->32->64
    run_layer<2,   128>(x_in, W1, b1, srcp, dstp, dinv, dinv2, bufH, bufX, N, E, stream);
    run_layer<128, 128>(bufX, W2, b2, srcp, dstp, dinv, dinv2, bufH, bufX, N, E, stream);
    run_layer<128, 32 >(bufX, W3, b3, srcp, dstp, dinv, dinv2, bufH, bufX, N, E, stream);
    run_layer<32,  32 >(bufX, W4, b4, srcp, dstp, dinv, dinv2, bufH, bufX, N, E, stream);
    run_layer<32,  64 >(bufX, W5, b5, srcp, dstp, dinv, dinv2, bufH, bufX, N, E, stream);

    // final linear 64 -> 1
    final_linear_kernel<<<(N + TB - 1) / TB, TB, 0, stream>>>(bufX, Wl, bl,
                                                              (float*)d_out, N, Od);
}